// DNC_83159156785506
// MI455X (gfx1250) — compile-verified
//
#include <hip/hip_runtime.h>
#include <math.h>

#define DEV __device__ __forceinline__

// ---------------- problem constants ----------------
constexpr int T = 32, BB = 16, X = 128, H = 512, NN = 512, W = 64, R = 4;
constexpr int H4  = 4 * H;          // 2048
constexpr int KIN = X + R * W;      // 384
constexpr int KTOT = KIN + H;       // 896
constexpr int CTRL = R*W + R + 3*R + W + 1 + W + W + R + 2;  // 471
constexpr float EPS = 1e-8f;

// ctrl vector layout (matches reference parse order)
constexpr int O_RK = 0;             // R*W raw read keys
constexpr int O_RB = R * W;         // R   read betas (1+softplus)
constexpr int O_RM = O_RB + R;      // 3R  read-mode logits (softmax over 3)
constexpr int O_WK = O_RM + 3 * R;  // W   write key (raw)
constexpr int O_WB = O_WK + W;      // 1   write beta (1+softplus)
constexpr int O_ER = O_WB + 1;      // W   erase (sigmoid)
constexpr int O_AD = O_ER + W;      // W   add (raw)
constexpr int O_FG = O_AD + W;      // R   free gates (sigmoid)
constexpr int O_AG = O_FG + R;      // 1   alloc gate (sigmoid)
constexpr int O_WG = O_AG + 1;      // 1   write gate (sigmoid)

// ---------------- workspace layout (floats) ----------------
constexpr size_t OFF_H    = 0;                        // B*H
constexpr size_t OFF_C    = OFF_H    + (size_t)BB*H;  // B*H
constexpr size_t OFF_M    = OFF_C    + (size_t)BB*H;  // B*N*W
constexpr size_t OFF_U    = OFF_M    + (size_t)BB*NN*W;
constexpr size_t OFF_L    = OFF_U    + (size_t)BB*NN;      // B*N*N
constexpr size_t OFF_P    = OFF_L    + (size_t)BB*NN*NN;
constexpr size_t OFF_WR   = OFF_P    + (size_t)BB*NN;      // B*R*N
constexpr size_t OFF_RV   = OFF_WR   + (size_t)BB*R*NN;    // B*R*W
constexpr size_t OFF_WW   = OFF_RV   + (size_t)BB*R*W;     // B*N
constexpr size_t STATE_TOTAL = OFF_WW + (size_t)BB*NN;     // zero/seed on init
constexpr size_t OFF_G    = STATE_TOTAL;                   // B*4H
constexpr size_t OFF_CT   = OFF_G    + (size_t)BB*H4;      // B*CTRL
constexpr size_t OFF_WS   = OFF_CT   + (size_t)BB*CTRL;    // B (ww sums)
constexpr size_t OFF_FW   = OFF_WS   + (size_t)BB;         // B*R*N
constexpr size_t OFF_BW   = OFF_FW   + (size_t)BB*R*NN;    // B*R*N
constexpr size_t OFF_IN   = OFF_BW   + (size_t)BB*R*NN;    // B*KTOT packed LSTM input

// ---------------- helpers ----------------
DEV float sigm(float x)      { return 1.f / (1.f + expf(-x)); }
DEV float softplusf(float x) { return (x > 20.f) ? x : log1pf(expf(x)); }

DEV float blk_max(float v, float* buf) {
    int n = threadIdx.x;
    buf[n] = v; __syncthreads();
    for (int s = blockDim.x >> 1; s > 0; s >>= 1) {
        if (n < s) buf[n] = fmaxf(buf[n], buf[n + s]);
        __syncthreads();
    }
    float r = buf[0]; __syncthreads(); return r;
}
DEV float blk_sum(float v, float* buf) {
    int n = threadIdx.x;
    buf[n] = v; __syncthreads();
    for (int s = blockDim.x >> 1; s > 0; s >>= 1) {
        if (n < s) buf[n] += buf[n + s];
        __syncthreads();
    }
    float r = buf[0]; __syncthreads(); return r;
}

// ---------------- WMMA types (fp32, 16x16x4) ----------------
typedef float v2f __attribute__((ext_vector_type(2)));
typedef float v8f __attribute__((ext_vector_type(8)));

// ===================================================================
// init: h = tile(h0), c = tile(c0), rest of persistent state = 0
// ===================================================================
__global__ void dnc_init(float* ws, const float* __restrict__ h0,
                         const float* __restrict__ c0) {
    size_t idx = (size_t)blockIdx.x * 256 + threadIdx.x;
    if (idx >= STATE_TOTAL) return;
    if (idx < OFF_C)       ws[idx] = h0[idx % H];   // OFF_C is a multiple of H
    else if (idx < OFF_M)  ws[idx] = c0[idx % H];
    else                   ws[idx] = 0.f;
}

// ===================================================================
// pack LSTM A operand: inp[b][0:896] = [x_t(128) | rv(256) | h(512)]
// makes the GEMM A fetch a single contiguous b64 per lane per K-step
// ===================================================================
__global__ void dnc_pack(const float* __restrict__ xt, const float* __restrict__ rv,
                         const float* __restrict__ h, float* __restrict__ inp) {
    int idx = blockIdx.x * 256 + threadIdx.x;
    if (idx >= BB * KTOT) return;
    int m = idx / KTOT, k = idx % KTOT;
    float v;
    if (k < X)        v = xt[m * X + k];
    else if (k < KIN) v = rv[m * (R * W) + (k - X)];
    else              v = h[m * H + (k - KIN)];
    inp[idx] = v;
}

// ===================================================================
// LSTM gates GEMM: g = inp @ [wx;wh] + b   (M=16, N=2048, K=896)
// one wave per 64-wide N slab (4 WMMA tiles); branch-free split K loops
// ===================================================================
__global__ void dnc_lstm_gemm(const float* __restrict__ inp, const float* __restrict__ wx,
                              const float* __restrict__ wh, const float* __restrict__ bias,
                              float* __restrict__ g) {
    const int lane  = threadIdx.x;
    const int m     = lane & 15;
    const int half  = lane >> 4;
    const int nbase = blockIdx.x * 64 + m;
    const float* arow = inp + m * KTOT + half * 2;
    v8f acc0 = {}, acc1 = {}, acc2 = {}, acc3 = {};

    // phase 1: K in [0, KIN) against lstm_wx
    for (int k0 = 0; k0 < KIN; k0 += 4) {
        const v2f a = *(const v2f*)(arow + k0);
        const float* w0 = wx + (size_t)(k0 + half * 2) * H4 + nbase;
        v2f b0, b1, b2, b3;
        b0[0] = w0[0];   b0[1] = w0[H4];
        b1[0] = w0[16];  b1[1] = w0[H4 + 16];
        b2[0] = w0[32];  b2[1] = w0[H4 + 32];
        b3[0] = w0[48];  b3[1] = w0[H4 + 48];
        acc0 = __builtin_amdgcn_wmma_f32_16x16x4_f32(false, a, false, b0, (short)0, acc0, false, false);
        acc1 = __builtin_amdgcn_wmma_f32_16x16x4_f32(false, a, false, b1, (short)0, acc1, false, false);
        acc2 = __builtin_amdgcn_wmma_f32_16x16x4_f32(false, a, false, b2, (short)0, acc2, false, false);
        acc3 = __builtin_amdgcn_wmma_f32_16x16x4_f32(false, a, false, b3, (short)0, acc3, false, false);
    }
    // phase 2: K in [KIN, KIN+H) against lstm_wh
    const float* arow2 = arow + KIN;
    for (int k0 = 0; k0 < H; k0 += 4) {
        const v2f a = *(const v2f*)(arow2 + k0);
        const float* w0 = wh + (size_t)(k0 + half * 2) * H4 + nbase;
        v2f b0, b1, b2, b3;
        b0[0] = w0[0];   b0[1] = w0[H4];
        b1[0] = w0[16];  b1[1] = w0[H4 + 16];
        b2[0] = w0[32];  b2[1] = w0[H4 + 32];
        b3[0] = w0[48];  b3[1] = w0[H4 + 48];
        acc0 = __builtin_amdgcn_wmma_f32_16x16x4_f32(false, a, false, b0, (short)0, acc0, false, false);
        acc1 = __builtin_amdgcn_wmma_f32_16x16x4_f32(false, a, false, b1, (short)0, acc1, false, false);
        acc2 = __builtin_amdgcn_wmma_f32_16x16x4_f32(false, a, false, b2, (short)0, acc2, false, false);
        acc3 = __builtin_amdgcn_wmma_f32_16x16x4_f32(false, a, false, b3, (short)0, acc3, false, false);
    }

    v8f accs[4] = {acc0, acc1, acc2, acc3};
#pragma unroll
    for (int tile = 0; tile < 4; ++tile) {
        const int ncol = nbase + tile * 16;
#pragma unroll
        for (int v = 0; v < 8; ++v) {
            const int row = v + 8 * half;
            g[(size_t)row * H4 + ncol] = accs[tile][v] + bias[ncol];
        }
    }
}

// ===================================================================
// LSTM activations: c,h update
// ===================================================================
__global__ void dnc_lstm_act(const float* __restrict__ g, float* h, float* c) {
    int idx = blockIdx.x * 256 + threadIdx.x;
    if (idx >= BB * H) return;
    int b = idx / H, j = idx % H;
    const float* gb = g + (size_t)b * H4;
    float gi = gb[j], gf = gb[H + j], gg = gb[2 * H + j], go = gb[3 * H + j];
    float cn = sigm(gf) * c[idx] + sigm(gi) * tanhf(gg);
    c[idx] = cn;
    h[idx] = sigm(go) * tanhf(cn);
}

// ===================================================================
// controller GEMM: ctrl = h @ w_ctrl + b_ctrl  (M=16, N=471, K=512)
// 2 tiles/wave; clamped column for branch-free in-bounds loads
// ===================================================================
__global__ void dnc_ctrl_gemm(const float* __restrict__ h, const float* __restrict__ wc,
                              const float* __restrict__ bc, float* __restrict__ ctrl) {
    const int lane  = threadIdx.x;
    const int m     = lane & 15;
    const int half  = lane >> 4;
    const int n0    = blockIdx.x * 32 + m;        // tile0 column
    const int n1    = n0 + 16;                    // tile1 column
    const int l0    = (n0 < CTRL) ? n0 : (CTRL - 1);   // clamped (always loadable)
    const int l1    = (n1 < CTRL) ? n1 : (CTRL - 1);
    const float* arow = h + m * H + half * 2;
    v8f acc0 = {}, acc1 = {};
    for (int k0 = 0; k0 < H; k0 += 4) {
        const v2f a = *(const v2f*)(arow + k0);
        const float* w0 = wc + (size_t)(k0 + half * 2) * CTRL;
        v2f b0, b1;
        b0[0] = w0[l0]; b0[1] = w0[CTRL + l0];
        b1[0] = w0[l1]; b1[1] = w0[CTRL + l1];
        acc0 = __builtin_amdgcn_wmma_f32_16x16x4_f32(false, a, false, b0, (short)0, acc0, false, false);
        acc1 = __builtin_amdgcn_wmma_f32_16x16x4_f32(false, a, false, b1, (short)0, acc1, false, false);
    }
    if (n0 < CTRL) {
#pragma unroll
        for (int v = 0; v < 8; ++v)
            ctrl[(size_t)(v + 8 * half) * CTRL + n0] = acc0[v] + bc[n0];
    }
    if (n1 < CTRL) {
#pragma unroll
        for (int v = 0; v < 8; ++v)
            ctrl[(size_t)(v + 8 * half) * CTRL + n1] = acc1[v] + bc[n1];
    }
}

// ===================================================================
// allocation + write weighting: cw (cosine softmax), psi, u update,
// bitonic argsort of u, exclusive cumprod, a, ww, sum(ww)
// one block (512 threads) per batch element
// ===================================================================
__global__ void dnc_alloc_write(const float* __restrict__ M, const float* __restrict__ ctrl,
                                const float* __restrict__ wr, float* u, float* ww,
                                float* wwsum) {
    const int b = blockIdx.x;
    const int n = threadIdx.x;
    __shared__ float sk[NN];
    __shared__ int   si[NN];
    __shared__ float sa[NN];
    __shared__ float sr[NN];
    const float* cb = ctrl + (size_t)b * CTRL;

    // cosine content score vs write key
    float wk2 = 0.f, dot = 0.f, m2 = 0.f;
    const float* Mr = M + ((size_t)b * NN + n) * W;
    for (int w = 0; w < W; ++w) {
        float wkv = cb[O_WK + w], mv = Mr[w];
        wk2 += wkv * wkv; dot += mv * wkv; m2 += mv * mv;
    }
    const float wb = 1.f + softplusf(cb[O_WB]);
    const float score = dot / (sqrtf(wk2) * sqrtf(m2) + EPS) * wb;
    const float mx = blk_max(score, sr);
    const float e  = expf(score - mx);
    const float cw = e / blk_sum(e, sr);

    // retention psi and usage update (uses previous-step wr and ww)
    float psi = 1.f;
#pragma unroll
    for (int r = 0; r < R; ++r)
        psi *= (1.f - sigm(cb[O_FG + r]) * wr[((size_t)b * R + r) * NN + n]);
    const float uo = u[b * NN + n];
    const float wp = ww[b * NN + n];
    const float un = (uo + wp - uo * wp) * psi;
    u[b * NN + n] = un;

    // ascending bitonic sort of (u, index)
    sk[n] = un; si[n] = n;
    __syncthreads();
    for (int ks = 2; ks <= NN; ks <<= 1)
        for (int j = ks >> 1; j > 0; j >>= 1) {
            const int ixj = n ^ j;
            if (ixj > n) {
                const bool up = ((n & ks) == 0);
                float kn = sk[n], kx = sk[ixj];
                if ((kn > kx) == up) {
                    sk[n] = kx; sk[ixj] = kn;
                    int t = si[n]; si[n] = si[ixj]; si[ixj] = t;
                }
            }
            __syncthreads();
        }

    // inclusive product scan of sorted usage -> exclusive cumprod
    sr[n] = sk[n];
    __syncthreads();
    for (int off = 1; off < NN; off <<= 1) {
        const float v = (n >= off) ? sr[n - off] : 1.f;
        __syncthreads();
        sr[n] *= v;
        __syncthreads();
    }
    const float cp = (n == 0) ? 1.f : sr[n - 1];
    sa[si[n]] = (1.f - sk[n]) * cp;    // scatter back to original slot order
    __syncthreads();

    const float ag = sigm(cb[O_AG]);
    const float wg = sigm(cb[O_WG]);
    const float wn = wg * (ag * sa[n] + (1.f - ag) * cw);
    ww[b * NN + n] = wn;
    const float s = blk_sum(wn, sr);
    if (n == 0) wwsum[b] = s;
}

// ===================================================================
// memory write: M = M*(1 - ww (x) erase) + ww (x) add
// ===================================================================
__global__ void dnc_mem(const float* __restrict__ ctrl, const float* __restrict__ ww,
                        float* M) {
    int idx = blockIdx.x * 256 + threadIdx.x;
    if (idx >= BB * NN * W) return;
    int b = idx / (NN * W), rem = idx % (NN * W), n = rem / W, w = rem % W;
    const float* cb = ctrl + (size_t)b * CTRL;
    const float wwv = ww[b * NN + n];
    const float er  = sigm(cb[O_ER + w]);
    const float ad  = cb[O_AD + w];
    M[idx] = M[idx] * (1.f - wwv * er) + wwv * ad;
}

// ===================================================================
// link matrix update (uses pre-update precedence p); zero diagonal
// ===================================================================
__global__ void dnc_link(const float* __restrict__ ww, const float* __restrict__ p,
                         float* L) {
    int idx = blockIdx.x * 256 + threadIdx.x;
    if (idx >= BB * NN * NN) return;
    int b = idx / (NN * NN), rem = idx % (NN * NN), i = rem / NN, j = rem % NN;
    if (i == j) { L[idx] = 0.f; return; }
    const float wi = ww[b * NN + i], wj = ww[b * NN + j];
    L[idx] = (1.f - wi - wj) * L[idx] + wi * p[b * NN + j];
}

// precedence update
__global__ void dnc_prec(const float* __restrict__ ww, const float* __restrict__ wwsum,
                         float* p) {
    int idx = blockIdx.x * 256 + threadIdx.x;
    if (idx >= BB * NN) return;
    int b = idx / NN;
    p[idx] = (1.f - wwsum[b]) * p[idx] + ww[idx];
}

// ===================================================================
// fw[b,r,i] = sum_j L[b,i,j] * wr[b,r,j]  — one 64-thread block per (b,i)
// ===================================================================
__global__ void dnc_fw(const float* __restrict__ L, const float* __restrict__ wr,
                       float* fw) {
    const int b = blockIdx.x / NN, i = blockIdx.x % NN;
    const int t = threadIdx.x;  // 64
    __shared__ float red[R][64];
    float acc[R] = {0.f, 0.f, 0.f, 0.f};
    const float* Lr = L + ((size_t)b * NN + i) * NN;
    for (int j = t; j < NN; j += 64) {
        const float lv = Lr[j];
#pragma unroll
        for (int r = 0; r < R; ++r) acc[r] += lv * wr[((size_t)b * R + r) * NN + j];
    }
#pragma unroll
    for (int r = 0; r < R; ++r) red[r][t] = acc[r];
    __syncthreads();
    if (t < R) {
        float s = 0.f;
        for (int q = 0; q < 64; ++q) s += red[t][q];
        fw[((size_t)b * R + t) * NN + i] = s;
    }
}

// bw[b,r,j] = sum_i L[b,i,j] * wr[b,r,i]  — one 512-thread block per batch
__global__ void dnc_bw(const float* __restrict__ L, const float* __restrict__ wr,
                       float* bw) {
    const int b = blockIdx.x;
    const int j = threadIdx.x;  // 512, coalesced along columns
    float acc[R] = {0.f, 0.f, 0.f, 0.f};
    for (int i = 0; i < NN; ++i) {
        const float lv = L[((size_t)b * NN + i) * NN + j];
#pragma unroll
        for (int r = 0; r < R; ++r) acc[r] += lv * wr[((size_t)b * R + r) * NN + i];
    }
#pragma unroll
    for (int r = 0; r < R; ++r) bw[((size_t)b * R + r) * NN + j] = acc[r];
}

// ===================================================================
// read heads: content softmax (post-write M), mode mix, wr update, rv = wr @ M
// one 512-thread block per (b, r)
// ===================================================================
__global__ void dnc_read(const float* __restrict__ M, const float* __restrict__ ctrl,
                         const float* __restrict__ fw, const float* __restrict__ bw,
                         float* wr, float* rv) {
    const int b = blockIdx.x / R, r = blockIdx.x % R;
    const int n = threadIdx.x;
    __shared__ float sr[NN];
    __shared__ float swr[NN];
    const float* cb = ctrl + (size_t)b * CTRL;

    float rk2 = 0.f, dot = 0.f, m2 = 0.f;
    const float* Mr = M + ((size_t)b * NN + n) * W;
    for (int w = 0; w < W; ++w) {
        float rkv = cb[O_RK + r * W + w], mv = Mr[w];
        rk2 += rkv * rkv; dot += mv * rkv; m2 += mv * mv;
    }
    const float rb = 1.f + softplusf(cb[O_RB + r]);
    const float score = dot / (sqrtf(rk2) * sqrtf(m2) + EPS) * rb;
    const float mx = blk_max(score, sr);
    const float e  = expf(score - mx);
    const float cr = e / blk_sum(e, sr);

    // read-mode softmax over {backward, content, forward}
    const float e0 = cb[O_RM + r * 3 + 0], e1 = cb[O_RM + r * 3 + 1], e2 = cb[O_RM + r * 3 + 2];
    const float mm = fmaxf(e0, fmaxf(e1, e2));
    const float x0 = expf(e0 - mm), x1 = expf(e1 - mm), x2 = expf(e2 - mm);
    const float inv = 1.f / (x0 + x1 + x2);

    const size_t idx = ((size_t)b * R + r) * NN + n;
    const float wnv = (x0 * inv) * bw[idx] + (x1 * inv) * cr + (x2 * inv) * fw[idx];
    wr[idx] = wnv;
    swr[n] = wnv;
    __syncthreads();

    if (n < W) {
        float acc = 0.f;
        for (int q = 0; q < NN; ++q) acc += swr[q] * M[((size_t)b * NN + q) * W + n];
        rv[((size_t)b * R + r) * W + n] = acc;
    }
}

// ===================================================================
// output GEMM: y_t = [h, rv] @ w_out + b_out  (M=16, N=128, K=768)
// 4 tiles/wave; split K loops (h part, rv part), branch-free
// ===================================================================
__global__ void dnc_out_gemm(const float* __restrict__ h, const float* __restrict__ rv,
                             const float* __restrict__ wo, const float* __restrict__ bo,
                             float* __restrict__ y) {
    const int lane  = threadIdx.x;
    const int m     = lane & 15;
    const int half  = lane >> 4;
    const int nbase = blockIdx.x * 64 + m;
    v8f acc0 = {}, acc1 = {}, acc2 = {}, acc3 = {};

    const float* arow = h + m * H + half * 2;
    for (int k0 = 0; k0 < H; k0 += 4) {
        const v2f a = *(const v2f*)(arow + k0);
        const float* w0 = wo + (size_t)(k0 + half * 2) * X + nbase;
        v2f b0, b1, b2, b3;
        b0[0] = w0[0];   b0[1] = w0[X];
        b1[0] = w0[16];  b1[1] = w0[X + 16];
        b2[0] = w0[32];  b2[1] = w0[X + 32];
        b3[0] = w0[48];  b3[1] = w0[X + 48];
        acc0 = __builtin_amdgcn_wmma_f32_16x16x4_f32(false, a, false, b0, (short)0, acc0, false, false);
        acc1 = __builtin_amdgcn_wmma_f32_16x16x4_f32(false, a, false, b1, (short)0, acc1, false, false);
        acc2 = __builtin_amdgcn_wmma_f32_16x16x4_f32(false, a, false, b2, (short)0, acc2, false, false);
        acc3 = __builtin_amdgcn_wmma_f32_16x16x4_f32(false, a, false, b3, (short)0, acc3, false, false);
    }
    const float* arow2 = rv + m * (R * W) + half * 2;
    for (int k0 = 0; k0 < R * W; k0 += 4) {
        const v2f a = *(const v2f*)(arow2 + k0);
        const float* w0 = wo + (size_t)(H + k0 + half * 2) * X + nbase;
        v2f b0, b1, b2, b3;
        b0[0] = w0[0];   b0[1] = w0[X];
        b1[0] = w0[16];  b1[1] = w0[X + 16];
        b2[0] = w0[32];  b2[1] = w0[X + 32];
        b3[0] = w0[48];  b3[1] = w0[X + 48];
        acc0 = __builtin_amdgcn_wmma_f32_16x16x4_f32(false, a, false, b0, (short)0, acc0, false, false);
        acc1 = __builtin_amdgcn_wmma_f32_16x16x4_f32(false, a, false, b1, (short)0, acc1, false, false);
        acc2 = __builtin_amdgcn_wmma_f32_16x16x4_f32(false, a, false, b2, (short)0, acc2, false, false);
        acc3 = __builtin_amdgcn_wmma_f32_16x16x4_f32(false, a, false, b3, (short)0, acc3, false, false);
    }

    v8f accs[4] = {acc0, acc1, acc2, acc3};
#pragma unroll
    for (int tile = 0; tile < 4; ++tile) {
        const int ncol = nbase + tile * 16;
#pragma unroll
        for (int v = 0; v < 8; ++v)
            y[(size_t)(v + 8 * half) * X + ncol] = accs[tile][v] + bo[ncol];
    }
}

// ===================================================================
// host launcher
// ===================================================================
extern "C" void kernel_launch(void* const* d_in, const int* in_sizes, int n_in,
                              void* d_out, int out_size, void* d_ws, size_t ws_size,
                              hipStream_t stream) {
    const float* x   = (const float*)d_in[0];
    const float* wx  = (const float*)d_in[1];
    const float* wh  = (const float*)d_in[2];
    const float* lb  = (const float*)d_in[3];
    const float* h0  = (const float*)d_in[4];
    const float* c0  = (const float*)d_in[5];
    const float* wc  = (const float*)d_in[6];
    const float* bc  = (const float*)d_in[7];
    const float* wo  = (const float*)d_in[8];
    const float* bo  = (const float*)d_in[9];
    float* out = (float*)d_out;

    float* ws   = (float*)d_ws;
    float* h    = ws + OFF_H;
    float* c    = ws + OFF_C;
    float* M    = ws + OFF_M;
    float* u    = ws + OFF_U;
    float* L    = ws + OFF_L;
    float* p    = ws + OFF_P;
    float* wr   = ws + OFF_WR;
    float* rv   = ws + OFF_RV;
    float* ww   = ws + OFF_WW;
    float* g    = ws + OFF_G;
    float* ct   = ws + OFF_CT;
    float* wwsum= ws + OFF_WS;
    float* fwb  = ws + OFF_FW;
    float* bwb  = ws + OFF_BW;
    float* inp  = ws + OFF_IN;

    dnc_init<<<(unsigned)((STATE_TOTAL + 255) / 256), 256, 0, stream>>>(ws, h0, c0);

    for (int t = 0; t < T; ++t) {
        const float* xt = x + (size_t)t * BB * X;
        float* yt = out + (size_t)t * BB * X;

        dnc_pack<<<(BB * KTOT + 255) / 256, 256, 0, stream>>>(xt, rv, h, inp);
        dnc_lstm_gemm<<<H4 / 64, 32, 0, stream>>>(inp, wx, wh, lb, g);
        dnc_lstm_act<<<(BB * H + 255) / 256, 256, 0, stream>>>(g, h, c);
        dnc_ctrl_gemm<<<15, 32, 0, stream>>>(h, wc, bc, ct);
        dnc_alloc_write<<<BB, NN, 0, stream>>>(M, ct, wr, u, ww, wwsum);
        dnc_mem<<<(BB * NN * W + 255) / 256, 256, 0, stream>>>(ct, ww, M);
        dnc_link<<<(BB * NN * NN + 255) / 256, 256, 0, stream>>>(ww, p, L);
        dnc_prec<<<(BB * NN + 255) / 256, 256, 0, stream>>>(ww, wwsum, p);
        dnc_fw<<<BB * NN, 64, 0, stream>>>(L, wr, fwb);
        dnc_bw<<<BB, NN, 0, stream>>>(L, wr, bwb);
        dnc_read<<<BB * R, NN, 0, stream>>>(M, ct, fwb, bwb, wr, rv);
        dnc_out_gemm<<<X / 64, 32, 0, stream>>>(h, rv, wo, bo, yt);
    }
}